// NetLinkTrain_41953240547886
// MI455X (gfx1250) — compile-verified
//
#include <hip/hip_runtime.h>

typedef __attribute__((ext_vector_type(2))) float v2f;
typedef __attribute__((ext_vector_type(8))) float v8f;

#define HID 128   // hidden dim (x, W1, W2 are 128-wide)

// ---------------- degree / normalization ----------------
__global__ void k_deg_init(float* deg, int n) {
  int i = blockIdx.x * blockDim.x + threadIdx.x;
  if (i < n) deg[i] = 1.0f;                    // self-loop contributes 1 at every node
}

__global__ void k_deg_accum(const int* __restrict__ dst, float* deg, int E) {
  int i = blockIdx.x * blockDim.x + threadIdx.x;
  if (i < E) atomicAdd(&deg[dst[i]], 1.0f);
}

__global__ void k_dinv(float* deg, int n) {
  int i = blockIdx.x * blockDim.x + threadIdx.x;
  if (i < n) { float d = deg[i]; deg[i] = d > 0.0f ? rsqrtf(d) : 0.0f; }
}

// ---------------- dense GEMM: Y[n,128] = X[n,128] @ W[128,128] ----------------
// One wave computes one 16x16 output tile via V_WMMA_F32_16X16X4_F32 (exact fp32).
// Block = 256 threads = 8 waves = 8 column tiles covering all 128 output cols.
__global__ __launch_bounds__(256) void k_gemm128(const float* __restrict__ X,
                                                 const float* __restrict__ W,
                                                 float* __restrict__ Y, int nrows) {
  const int lane = threadIdx.x & 31;
  const int wv   = threadIdx.x >> 5;        // 0..7 : which 16-col tile
  const int row_base = blockIdx.x << 4;     // 16 rows per block
  if (row_base >= nrows) return;
  const int m     = lane & 15;
  const int khalf = (lane >> 4) << 1;       // lanes 0-15 -> K+0/K+1, lanes 16-31 -> K+2/K+3
  const int col   = (wv << 4) + m;
  const float* xr = X + (long)(row_base + m) * HID;
  v8f c = {};
  for (int k0 = 0; k0 < HID; k0 += 4) {
    v2f a, b;
    a.x = xr[k0 + khalf];                               // A: 16x4, row m
    a.y = xr[k0 + khalf + 1];
    b.x = W[(long)(k0 + khalf) * HID + col];            // B: 4x16, col `col`
    b.y = W[(long)(k0 + khalf + 1) * HID + col];
    c = __builtin_amdgcn_wmma_f32_16x16x4_f32(false, a, false, b, (short)0, c,
                                              false, false);
  }
  // C/D layout: VGPR r -> M = r (lanes 0-15) or r+8 (lanes 16-31), N = lane%16
  const int rtop = row_base + ((lane >> 4) << 3);
  for (int r = 0; r < 8; ++r)
    Y[(long)(rtop + r) * HID + col] = c[r];
}

// ---------------- self-loop term: out[i,:] = dinv[i]^2 * xw[i,:] ----------------
__global__ void k_selfloop_init(const float* __restrict__ xw,
                                const float* __restrict__ dinv,
                                float* __restrict__ out, int total) {
  int i = blockIdx.x * blockDim.x + threadIdx.x;
  if (i < total) {
    float d = dinv[i >> 7];                 // node = i / 128
    out[i] = d * d * xw[i];
  }
}

// ---------------- edge scatter: out[dst,:] += dinv[src]*dinv[dst] * xw[src,:] ----------------
// One wave per edge; each lane moves a float4 (32 lanes * 4 = 128 features).
__global__ __launch_bounds__(256) void k_edge_scatter(const float* __restrict__ xw,
                                                      const float* __restrict__ dinv,
                                                      const int* __restrict__ src,
                                                      const int* __restrict__ dst,
                                                      float* __restrict__ out, int E) {
  int gw   = (blockIdx.x * blockDim.x + threadIdx.x) >> 5;
  int lane = threadIdx.x & 31;
  if (gw >= E) return;
  int s = src[gw];
  int d = dst[gw];
  float nrm = dinv[s] * dinv[d];
  const float4* ps = (const float4*)(xw + (long)s * HID);
  float4 v = ps[lane];
  float* po = out + (long)d * HID + lane * 4;
  atomicAdd(po + 0, nrm * v.x);
  atomicAdd(po + 1, nrm * v.y);
  atomicAdd(po + 2, nrm * v.z);
  atomicAdd(po + 3, nrm * v.w);
}

__global__ void k_relu(float* z, int total) {
  int i = blockIdx.x * blockDim.x + threadIdx.x;
  if (i < total) z[i] = fmaxf(z[i], 0.0f);
}

// ---------------- decoder: out[j,:2] = [z[a]|z[b]] @ Wlin^T ----------------
// One wave per (pos|neg) edge, butterfly reduce over 32 lanes (wave32).
__global__ __launch_bounds__(256) void k_decode(const float* __restrict__ z,
                                                const int* __restrict__ pe,
                                                const int* __restrict__ ne,
                                                const float* __restrict__ Wlin,
                                                float* __restrict__ out,
                                                int Ep, int En) {
  int gw   = (blockIdx.x * blockDim.x + threadIdx.x) >> 5;
  int lane = threadIdx.x & 31;
  int total = Ep + En;
  if (gw >= total) return;
  int a, b;
  if (gw < Ep) { a = pe[gw];       b = pe[Ep + gw]; }
  else         { int j = gw - Ep;  a = ne[j]; b = ne[En + j]; }
  const float* za = z + (long)a * HID;
  const float* zb = z + (long)b * HID;
  float s0 = 0.0f, s1 = 0.0f;
  for (int k = lane; k < HID; k += 32) {
    float va = za[k], vb = zb[k];
    s0 += va * Wlin[k]       + vb * Wlin[HID + k];        // row 0 of Wlin[2,256]
    s1 += va * Wlin[256 + k] + vb * Wlin[256 + HID + k];  // row 1
  }
  for (int off = 16; off > 0; off >>= 1) {
    s0 += __shfl_xor(s0, off, 32);
    s1 += __shfl_xor(s1, off, 32);
  }
  if (lane == 0) {
    out[(long)gw * 2 + 0] = s0;
    out[(long)gw * 2 + 1] = s1;
  }
}

extern "C" void kernel_launch(void* const* d_in, const int* in_sizes, int n_in,
                              void* d_out, int out_size, void* d_ws, size_t ws_size,
                              hipStream_t stream) {
  const float* x    = (const float*)d_in[0];
  const int*   ei   = (const int*)d_in[1];
  const int*   pe   = (const int*)d_in[2];
  const int*   ne   = (const int*)d_in[3];
  const float* W1   = (const float*)d_in[4];
  const float* W2   = (const float*)d_in[5];
  const float* Wlin = (const float*)d_in[6];
  float* out = (float*)d_out;

  const int N  = in_sizes[0] / HID;   // 50000 nodes
  const int E  = in_sizes[1] / 2;     // 600000 edges
  const int Ep = in_sizes[2] / 2;     // 100000 pos
  const int En = in_sizes[3] / 2;     // 100000 neg
  const int* src = ei;
  const int* dst = ei + E;

  // workspace layout: dinv | bufA (N*128) | bufB (N*128)
  char*  ws   = (char*)d_ws;
  float* dinv = (float*)ws;
  size_t offA = (((size_t)N * 4) + 255) & ~(size_t)255;
  float* bufA = (float*)(ws + offA);
  float* bufB = bufA + (size_t)N * HID;

  const int BT = 256;
  int totalF = N * HID;

  // 1) symmetric normalization
  k_deg_init <<<(N + BT - 1) / BT, BT, 0, stream>>>(dinv, N);
  k_deg_accum<<<(E + BT - 1) / BT, BT, 0, stream>>>(dst, dinv, E);
  k_dinv     <<<(N + BT - 1) / BT, BT, 0, stream>>>(dinv, N);

  // 2) conv1: bufA = x @ W1 ; bufB = selfloop + scatter ; relu
  k_gemm128      <<<N / 16, 256, 0, stream>>>(x, W1, bufA, N);
  k_selfloop_init<<<(totalF + BT - 1) / BT, BT, 0, stream>>>(bufA, dinv, bufB, totalF);
  k_edge_scatter <<<(E + 7) / 8, 256, 0, stream>>>(bufA, dinv, src, dst, bufB, E);
  k_relu         <<<(totalF + BT - 1) / BT, BT, 0, stream>>>(bufB, totalF);

  // 3) conv2: bufA = relu(z1) @ W2 ; bufB = selfloop + scatter (bufB reused as z2)
  k_gemm128      <<<N / 16, 256, 0, stream>>>(bufB, W2, bufA, N);
  k_selfloop_init<<<(totalF + BT - 1) / BT, BT, 0, stream>>>(bufA, dinv, bufB, totalF);
  k_edge_scatter <<<(E + 7) / 8, 256, 0, stream>>>(bufA, dinv, src, dst, bufB, E);

  // 4) decode pos+neg edges
  int totE = Ep + En;
  k_decode<<<(totE + 7) / 8, 256, 0, stream>>>(bufB, pe, ne, Wlin, out, Ep, En);
}